// Decoder_78280073937032
// MI455X (gfx1250) — compile-verified
//
#include <hip/hip_runtime.h>
#include <math.h>

#define BB 256
#define NN 2000
#define DD 128
#define HH 8
#define EE 16
#define FD 512   // 4*D

typedef float v2f __attribute__((ext_vector_type(2)));
typedef float v8f __attribute__((ext_vector_type(8)));

// Only meaningful in the device pass; the host pass (x86) doesn't declare
// AMDGCN builtins, which is what tripped round 1.
#if defined(__HIP_DEVICE_COMPILE__)
#if !__has_builtin(__builtin_amdgcn_wmma_f32_16x16x4_f32)
#error "DEVICE pass: __builtin_amdgcn_wmma_f32_16x16x4_f32 not available"
#endif
#endif

// D = A(16x4) * B(4x16) + C, fp32. Wrapped so the host pass never sees the
// target builtin (kernel bodies are semantic-checked on both passes).
__device__ __forceinline__ v8f wmma_f32_16x16x4(v2f a, v2f b, v8f c) {
#if defined(__HIP_DEVICE_COMPILE__)
  return __builtin_amdgcn_wmma_f32_16x16x4_f32(false, a, false, b,
                                               (short)0, c, false, false);
#else
  (void)a; (void)b;
  return c;
#endif
}

// ---------------------------------------------------------------------------
// K1: per-batch prep. context -> x -> LSTM(h0=c0=0) -> h1 -> q -> fold q into
// Wk producing A[b] = [H, D], pre-scaled by 1/sqrt(E) = 0.25.
// One block (128 threads) per batch.
// ---------------------------------------------------------------------------
__global__ void __launch_bounds__(128)
prep_kernel(const float* __restrict__ emb, const float* __restrict__ gemb,
            const float* __restrict__ semb, const float* __restrict__ Wf,
            const float* __restrict__ bf, const float* __restrict__ Wih,
            const float* __restrict__ bih, const float* __restrict__ bhh,
            const float* __restrict__ Wq, const float* __restrict__ Wk,
            const int* __restrict__ firstn, const int* __restrict__ curn,
            float* __restrict__ Af) {
  const int b = blockIdx.x;
  const int tid = threadIdx.x;            // 0..127
  __shared__ float ctxv[FD];
  __shared__ float xL[DD];
  __shared__ float h1L[DD];
  __shared__ float qL[DD];

  const int fn = firstn[b], cn = curn[b];
  ctxv[tid]          = gemb[b * DD + tid];
  ctxv[DD + tid]     = emb[((size_t)b * NN + fn) * DD + tid];
  ctxv[2 * DD + tid] = emb[((size_t)b * NN + cn) * DD + tid];
  ctxv[3 * DD + tid] = semb[b * DD + tid];
  __syncthreads();

  {
    float acc = bf[tid];
    const float* wrow = Wf + (size_t)tid * FD;
    for (int j = 0; j < FD; ++j) acc += wrow[j] * ctxv[j];
    xL[tid] = acc;
  }
  __syncthreads();

  {
    // gate order i,f,g,o; c0 = 0 so f-gate is dead.
    float ai = bih[tid] + bhh[tid];
    float ag = bih[2 * DD + tid] + bhh[2 * DD + tid];
    float ao = bih[3 * DD + tid] + bhh[3 * DD + tid];
    const float* wi = Wih + (size_t)tid * DD;
    const float* wg = Wih + (size_t)(2 * DD + tid) * DD;
    const float* wo = Wih + (size_t)(3 * DD + tid) * DD;
    for (int j = 0; j < DD; ++j) {
      const float xv = xL[j];
      ai += wi[j] * xv; ag += wg[j] * xv; ao += wo[j] * xv;
    }
    const float si = 1.0f / (1.0f + __expf(-ai));
    const float so = 1.0f / (1.0f + __expf(-ao));
    const float c1 = si * tanhf(ag);
    h1L[tid] = so * tanhf(c1);
  }
  __syncthreads();

  {
    float acc = 0.0f;
    const float* wq = Wq + (size_t)tid * DD;
    for (int j = 0; j < DD; ++j) acc += wq[j] * h1L[j];
    qL[tid] = acc;                        // q[d], d = e*H + h
  }
  __syncthreads();

  // A[b][h][j] = 0.25 * sum_e q[e*H+h] * Wk[e*H+h][j]
  for (int h = 0; h < HH; ++h) {
    float acc = 0.0f;
    for (int e = 0; e < EE; ++e)
      acc += qL[e * HH + h] * Wk[(size_t)(e * HH + h) * DD + tid];
    Af[((size_t)b * HH + h) * DD + tid] = 0.25f * acc;
  }
}

// ---------------------------------------------------------------------------
// K2: scores[b,n,h] = emb[b,n,:] . A[b,h,:]   (scale already folded into A)
// Streaming pass 1 over node_embeddings. WMMA f32 16x16x4:
//   A-frag = 16 nodes x 4 k ; B-frag = 4 k x 16 cols (cols 0..7 = heads).
// Block = 256 threads = 8 waves, each wave owns a 16-node tile -> 128 nodes.
// ---------------------------------------------------------------------------
__global__ void __launch_bounds__(256)
scores_kernel(const float* __restrict__ emb, const float* __restrict__ Af,
              float* __restrict__ scores) {
  const int b = blockIdx.y;
  const int tileBase = blockIdx.x * 128;
  const int tid = threadIdx.x;
  const int wave = tid >> 5;
  const int lane = tid & 31;
  const int lhalf = lane >> 4;            // K sub-chunk select
  const int l15 = lane & 15;              // M (A) / N (B) index

  __shared__ float sAf[HH * DD];          // 4 KB, L2-hot source
  for (int i = tid; i < HH * DD; i += 256) sAf[i] = Af[(size_t)b * HH * DD + i];
  __syncthreads();

  const int nodeBase = tileBase + wave * 16;
  int nodeA = nodeBase + l15;
  if (nodeA > NN - 1) nodeA = NN - 1;     // clamp so EXEC stays all-ones
  const float* rowA = emb + ((size_t)b * NN + nodeA) * DD;

  const float hm = (l15 < HH) ? 1.0f : 0.0f;
  const float* afp = &sAf[(l15 & 7) * DD];

  v8f c = {0.f, 0.f, 0.f, 0.f, 0.f, 0.f, 0.f, 0.f};
  #pragma unroll 8
  for (int k0 = 0; k0 < DD; k0 += 4) {
    const int kk = k0 + 2 * lhalf;
    v2f a = *reinterpret_cast<const v2f*>(rowA + kk);
    v2f bfr = *reinterpret_cast<const v2f*>(afp + kk);
    bfr[0] *= hm; bfr[1] *= hm;
    c = wmma_f32_16x16x4(a, bfr, c);
  }

  // D tile: vgpr r -> node = nodeBase + r + 8*lhalf, col = l15 (head if < 8)
  if (l15 < HH) {
    #pragma unroll
    for (int r = 0; r < 8; ++r) {
      const int node = nodeBase + r + 8 * lhalf;
      if (node < NN)
        scores[((size_t)b * NN + node) * HH + l15] = c[r];
    }
  }
}

// ---------------------------------------------------------------------------
// K3: masked softmax over n per (b,h). Wave h of each block handles head h.
// probs written in [B,H,N] layout for pass-2 A-fragment loads.
// ---------------------------------------------------------------------------
__global__ void __launch_bounds__(256)
softmax_kernel(const float* __restrict__ scores, const int* __restrict__ mask,
               float* __restrict__ probs) {
  const int b = blockIdx.x;
  const int h = threadIdx.x >> 5;
  const int lane = threadIdx.x & 31;
  const float NEGI = -__builtin_inff();

  float m = NEGI;
  for (int n = lane; n < NN; n += 32)
    if (mask[(size_t)b * NN + n] != 0)
      m = fmaxf(m, scores[((size_t)b * NN + n) * HH + h]);
  for (int off = 16; off; off >>= 1) m = fmaxf(m, __shfl_xor(m, off, 32));

  float sum = 0.0f;
  for (int n = lane; n < NN; n += 32) {
    float e = 0.0f;
    if (mask[(size_t)b * NN + n] != 0)
      e = __expf(scores[((size_t)b * NN + n) * HH + h] - m);
    sum += e;
  }
  for (int off = 16; off; off >>= 1) sum += __shfl_xor(sum, off, 32);
  const float inv = 1.0f / sum;

  for (int n = lane; n < NN; n += 32) {
    float e = 0.0f;
    if (mask[(size_t)b * NN + n] != 0)
      e = __expf(scores[((size_t)b * NN + n) * HH + h] - m) * inv;
    probs[((size_t)b * HH + h) * NN + n] = e;
  }
}

// ---------------------------------------------------------------------------
// K4: streaming pass 2: pooled[h,:] = sum_n probs[h,n] * emb[n,:]  (WMMA),
// then in-block: att -> ctx -> w[b], s[b].
// One block (8 waves) per batch; wave w owns 16-dim column tile c0 = 16*w.
// ---------------------------------------------------------------------------
__global__ void __launch_bounds__(256)
pooled_ctx_kernel(const float* __restrict__ emb, const float* __restrict__ probs,
                  const float* __restrict__ Wv, const float* __restrict__ Wproj,
                  const float* __restrict__ Wlk, const float* __restrict__ blk,
                  float* __restrict__ wvec, float* __restrict__ svals) {
  const int b = blockIdx.x;
  const int tid = threadIdx.x;
  const int wave = tid >> 5;
  const int lane = tid & 31;
  const int lhalf = lane >> 4;
  const int l15 = lane & 15;
  const int c0 = wave * 16;

  __shared__ float pooled[HH * DD];
  __shared__ float attL[DD];
  __shared__ float ctxL[DD];

  const float hm = (l15 < HH) ? 1.0f : 0.0f;   // pad heads 8..15 with zeros
  const float* pRow = probs + ((size_t)b * HH + (l15 & 7)) * NN + 2 * lhalf;
  const float* eCol = emb + (size_t)b * NN * DD + c0 + l15;

  v8f c = {0.f, 0.f, 0.f, 0.f, 0.f, 0.f, 0.f, 0.f};
  for (int n0 = 0; n0 < NN; n0 += 4) {         // 500 exact chunks
    __builtin_prefetch(eCol + (size_t)(n0 + 32 + 2 * lhalf) * DD, 0, 1);
    v2f a = *reinterpret_cast<const v2f*>(pRow + n0);
    a[0] *= hm; a[1] *= hm;
    v2f bfr;
    bfr[0] = eCol[(size_t)(n0 + 2 * lhalf) * DD];
    bfr[1] = eCol[(size_t)(n0 + 2 * lhalf + 1) * DD];
    c = wmma_f32_16x16x4(a, bfr, c);
  }
  // lanes 0..15 hold heads 0..7 in c[0..7]; column = c0 + l15
  if (lhalf == 0) {
    #pragma unroll
    for (int r = 0; r < 8; ++r) pooled[r * DD + c0 + l15] = c[r];
  }
  __syncthreads();

  if (tid < DD) {                              // att_flat[h*16+e], tid = h*16+e
    const int h = tid >> 4, e = tid & 15;
    const float* wv = Wv + (size_t)(e * HH + h) * DD;
    float acc = 0.0f;
    for (int d = 0; d < DD; ++d) acc += pooled[h * DD + d] * wv[d];
    attL[tid] = acc;
  }
  __syncthreads();
  if (tid < DD) {                              // ctx[d] = att . Wproj[d,:]
    const float* wp = Wproj + (size_t)tid * DD;
    float acc = 0.0f;
    for (int j = 0; j < DD; ++j) acc += attL[j] * wp[j];
    ctxL[tid] = acc;
  }
  __syncthreads();
  if (tid < DD) {                              // w[j] = sum_d ctx[d]*Wlk[d,j]
    float acc = 0.0f;
    for (int d = 0; d < DD; ++d) acc += ctxL[d] * Wlk[(size_t)d * DD + tid];
    wvec[(size_t)b * DD + tid] = acc;
  }
  if (tid == 0) {
    float s = 0.0f;
    for (int d = 0; d < DD; ++d) s += ctxL[d] * blk[d];
    svals[b] = s;
  }
}

// ---------------------------------------------------------------------------
// K5: streaming pass 3: comp[b,n] = mask?-inf : 10*tanh((emb.w + s)/sqrt(D)).
// One wave per node (float4-coalesced 512B row), shuffle reduction.
// Block covers 32 nodes (8 waves x 4 nodes).
// ---------------------------------------------------------------------------
__global__ void __launch_bounds__(256)
comp_kernel(const float* __restrict__ emb, const int* __restrict__ mask,
            const float* __restrict__ wvec, const float* __restrict__ svals,
            float* __restrict__ out) {
  const int b = blockIdx.y;
  const int tid = threadIdx.x;
  const int wave = tid >> 5;
  const int lane = tid & 31;
  __shared__ float wL[DD];
  __shared__ float sL;
  if (tid < DD) wL[tid] = wvec[(size_t)b * DD + tid];
  if (tid == 0) sL = svals[b];
  __syncthreads();
  const float s = sL;
  const float4 wv = *reinterpret_cast<const float4*>(&wL[4 * lane]);
  const int nodeBase = blockIdx.x * 32 + wave * 4;
  const float INVSQ = 0.08838834764831845f;    // 1/sqrt(128)

  for (int i = 0; i < 4; ++i) {
    const int node = nodeBase + i;
    if (node >= NN) break;                     // wave-uniform
    const float4 ev = *reinterpret_cast<const float4*>(
        emb + ((size_t)b * NN + node) * DD + 4 * lane);
    float dot = ev.x * wv.x + ev.y * wv.y + ev.z * wv.z + ev.w * wv.w;
    for (int off = 16; off; off >>= 1) dot += __shfl_xor(dot, off, 32);
    if (lane == 0) {
      const float v = 10.0f * tanhf((dot + s) * INVSQ);
      out[(size_t)b * NN + node] =
          (mask[(size_t)b * NN + node] == 1) ? -__builtin_inff() : v;
    }
  }
}

// ---------------------------------------------------------------------------
extern "C" void kernel_launch(void* const* d_in, const int* in_sizes, int n_in,
                              void* d_out, int out_size, void* d_ws, size_t ws_size,
                              hipStream_t stream) {
  const float* emb   = (const float*)d_in[0];   // [B,N,D]
  const float* gemb  = (const float*)d_in[1];   // [B,D]
  const float* semb  = (const float*)d_in[2];   // [B,D]
  const float* Wf    = (const float*)d_in[3];   // [D,4D]
  const float* bf    = (const float*)d_in[4];   // [D]
  const float* Wih   = (const float*)d_in[5];   // [4D,D]
  // d_in[6] = W_hh (unused: h0 = 0)
  const float* bih   = (const float*)d_in[7];   // [4D]
  const float* bhh   = (const float*)d_in[8];   // [4D]
  const float* Wq    = (const float*)d_in[9];   // [D,D]
  const float* Wk    = (const float*)d_in[10];  // [D,D]
  const float* Wv    = (const float*)d_in[11];  // [D,D]
  const float* Wproj = (const float*)d_in[12];  // [D,D]
  const float* Wlk   = (const float*)d_in[13];  // [D,D]
  const float* blk   = (const float*)d_in[14];  // [D]
  const int* firstn  = (const int*)d_in[15];    // [B]
  const int* curn    = (const int*)d_in[16];    // [B]
  const int* mask    = (const int*)d_in[17];    // [B,N]
  float* outp        = (float*)d_out;           // [B,N]

  float* ws = (float*)d_ws;
  float* Af     = ws;                                // B*H*D    =   262,144
  float* scores = Af + (size_t)BB * HH * DD;         // B*N*H    = 4,096,000
  float* probs  = scores + (size_t)BB * NN * HH;     // B*H*N    = 4,096,000
  float* wvec   = probs + (size_t)BB * HH * NN;      // B*D      =    32,768
  float* svals  = wvec + (size_t)BB * DD;            // B        =       256
  (void)ws_size; (void)in_sizes; (void)n_in; (void)out_size;

  prep_kernel<<<BB, 128, 0, stream>>>(emb, gemb, semb, Wf, bf, Wih, bih, bhh,
                                      Wq, Wk, firstn, curn, Af);
  scores_kernel<<<dim3(16, BB), 256, 0, stream>>>(emb, Af, scores);
  softmax_kernel<<<BB, 256, 0, stream>>>(scores, mask, probs);
  pooled_ctx_kernel<<<BB, 256, 0, stream>>>(emb, probs, Wv, Wproj, Wlk, blk,
                                            wvec, svals);
  comp_kernel<<<dim3(63, BB), 256, 0, stream>>>(emb, mask, wvec, svals, outp);
}